// MyMultiHeadAttention_19404662243357
// MI455X (gfx1250) — compile-verified
//
#include <hip/hip_runtime.h>
#include <math.h>

typedef __attribute__((ext_vector_type(2))) float v2f;
typedef __attribute__((ext_vector_type(8))) float v8f;

#define B_   2
#define L_   2048
#define D_   512
#define H_   8
#define DK_  64
#define TD_  1536
#define NEGV (-1e18f)
// 1/sqrt(512) — reference scales by sqrt(model_dim)
#define SCALE 0.04419417382415922f

static __device__ __forceinline__ v8f wmma4(v2f a, v2f b, v8f c) {
  // V_WMMA_F32_16X16X4_F32: (neg_a, A, neg_b, B, c_mod, C, reuse_a, reuse_b)
  return __builtin_amdgcn_wmma_f32_16x16x4_f32(false, a, false, b, (short)0, c,
                                               false, false);
}

// ---------------------------------------------------------------------------
// Kernel 1: qkv = q @ W_qkv + b_qkv, scattered into per-head Qw/Kw/Vw [B,H,L,DK]
// ---------------------------------------------------------------------------
__global__ __launch_bounds__(128) void qkv_kernel(
    const float* __restrict__ q, const float* __restrict__ Wqkv,
    const float* __restrict__ bqkv, float* __restrict__ Qw,
    float* __restrict__ Kw, float* __restrict__ Vw) {
  const int lane = threadIdx.x & 31;
  const int wave = threadIdx.x >> 5;
  const int half = lane >> 4;
  const int lc   = lane & 15;
  const int koff = half * 2;
  const int m0   = blockIdx.x * 64 + wave * 16;  // rows of [B*L, D]
  const int n0   = blockIdx.y * 64;              // cols of [*, 3D]
  const int arow = m0 + lc;

  v8f acc[4] = {};
  for (int kb = 0; kb < D_; kb += 4) {
    v2f a = *(const v2f*)&q[(size_t)arow * D_ + kb + koff];
#pragma unroll
    for (int t = 0; t < 4; ++t) {
      int col = n0 + t * 16 + lc;
      v2f b;
      b[0] = Wqkv[(size_t)(kb + koff) * TD_ + col];
      b[1] = Wqkv[(size_t)(kb + koff + 1) * TD_ + col];
      acc[t] = wmma4(a, b, acc[t]);
    }
  }
#pragma unroll
  for (int t = 0; t < 4; ++t) {
    int gcol = n0 + t * 16 + lc;
    float bias = bqkv[gcol];
    int region = gcol >> 9;       // 0:Q 1:K 2:V
    int hh     = (gcol & 511) >> 6;
    int dk     = gcol & 63;
    float* dst = (region == 0) ? Qw : (region == 1) ? Kw : Vw;
#pragma unroll
    for (int i = 0; i < 8; ++i) {
      int grow = m0 + i + half * 8;
      int bb   = grow >> 11;            // /L_
      int l    = grow & (L_ - 1);
      dst[(((size_t)(bb * H_ + hh)) * L_ + l) * DK_ + dk] = acc[t][i] + bias;
    }
  }
}

// ---------------------------------------------------------------------------
// Kernel 2: per (b,h) flash-style attention, writes normalized probs + ctx
// ---------------------------------------------------------------------------
__global__ __launch_bounds__(128) void attn_kernel(
    const float* __restrict__ Qw, const float* __restrict__ Kw,
    const float* __restrict__ Vw, const unsigned char* __restrict__ mask,
    float* __restrict__ attn_out, float* __restrict__ ctx) {
  __shared__ float lds_p[4][16 * 16];  // per-wave P staging for C->A relayout
  const int lane = threadIdx.x & 31;
  const int wave = threadIdx.x >> 5;
  const int half = lane >> 4;
  const int lc   = lane & 15;
  const int koff = half * 2;
  const int bh   = blockIdx.x;   // b*H + h
  const int b    = bh / H_;
  const int h    = bh % H_;
  const int q0   = blockIdx.y * 64 + wave * 16;

  const size_t headbase = (size_t)bh * L_ * DK_;

  // Q tile 16x64 as A-fragments (reused across all K tiles, both passes)
  v2f qf[16];
  {
    const float* qrowp = &Qw[headbase + (size_t)(q0 + lc) * DK_];
#pragma unroll
    for (int ks = 0; ks < 16; ++ks)
      qf[ks] = *(const v2f*)&qrowp[ks * 4 + koff];
  }

  float rm[8], rs[8];
#pragma unroll
  for (int i = 0; i < 8; ++i) { rm[i] = -INFINITY; rs[i] = 0.f; }

  // ---- Pass 1: online row max + sum of exp --------------------------------
  for (int kc = 0; kc < L_; kc += 16) {
    const float* krowp = &Kw[headbase + (size_t)(kc + lc) * DK_];
    __builtin_prefetch(krowp + 16 * DK_, 0, 1);
    v8f s = {};
#pragma unroll
    for (int ks = 0; ks < 16; ++ks) {
      v2f bfr = *(const v2f*)&krowp[ks * 4 + koff];
      s = wmma4(qf[ks], bfr, s);
    }
#pragma unroll
    for (int i = 0; i < 8; ++i) {
      int qrow = q0 + i + half * 8;
      unsigned char mb = mask[((size_t)b * L_ + qrow) * L_ + kc + lc];
      float v = mb ? NEGV : s[i] * SCALE;
      float tm = v;  // max over the 16-lane half (one C row)
      tm = fmaxf(tm, __shfl_xor(tm, 1, 32));
      tm = fmaxf(tm, __shfl_xor(tm, 2, 32));
      tm = fmaxf(tm, __shfl_xor(tm, 4, 32));
      tm = fmaxf(tm, __shfl_xor(tm, 8, 32));
      float mnew = fmaxf(rm[i], tm);
      float e = __expf(v - mnew);
      e += __shfl_xor(e, 1, 32);
      e += __shfl_xor(e, 2, 32);
      e += __shfl_xor(e, 4, 32);
      e += __shfl_xor(e, 8, 32);
      rs[i] = rs[i] * __expf(rm[i] - mnew) + e;
      rm[i] = mnew;
    }
  }
  float rinv[8];
#pragma unroll
  for (int i = 0; i < 8; ++i) rinv[i] = 1.f / rs[i];

  // ---- Pass 2: recompute S, emit P, accumulate O = P @ V ------------------
  v8f o[4] = {};
  float* lds = lds_p[wave];
  const size_t attn_base = ((size_t)(h * B_ + b)) * L_ * L_;
  for (int kc = 0; kc < L_; kc += 16) {
    const float* krowp = &Kw[headbase + (size_t)(kc + lc) * DK_];
    v8f s = {};
#pragma unroll
    for (int ks = 0; ks < 16; ++ks) {
      v2f bfr = *(const v2f*)&krowp[ks * 4 + koff];
      s = wmma4(qf[ks], bfr, s);
    }
#pragma unroll
    for (int i = 0; i < 8; ++i) {
      int qrow = q0 + i + half * 8;
      unsigned char mb = mask[((size_t)b * L_ + qrow) * L_ + kc + lc];
      float v = mb ? NEGV : s[i] * SCALE;
      float p = __expf(v - rm[i]) * rinv[i];
      attn_out[attn_base + (size_t)qrow * L_ + kc + lc] = p;
      lds[(i + half * 8) * 16 + lc] = p;
    }
    asm volatile("s_wait_dscnt 0" ::: "memory");  // intra-wave LDS RAW fence
#pragma unroll
    for (int ks2 = 0; ks2 < 4; ++ks2) {
      int kk = ks2 * 4;
      v2f a2 = *(const v2f*)&lds[lc * 16 + kk + koff];
      const float* vp = &Vw[headbase + (size_t)(kc + kk + koff) * DK_];
#pragma unroll
      for (int t = 0; t < 4; ++t) {
        v2f bfr;
        bfr[0] = vp[t * 16 + lc];
        bfr[1] = vp[DK_ + t * 16 + lc];
        o[t] = wmma4(a2, bfr, o[t]);
      }
    }
  }
  // ctx[b, l, h*64 + col] (head-concat layout for the projection GEMM)
#pragma unroll
  for (int t = 0; t < 4; ++t) {
#pragma unroll
    for (int i = 0; i < 8; ++i) {
      int l = q0 + i + half * 8;
      ctx[((size_t)b * L_ + l) * D_ + h * DK_ + t * 16 + lc] = o[t][i];
    }
  }
}

// ---------------------------------------------------------------------------
// Kernel 3: out = ctx @ W_proj + b_proj
// ---------------------------------------------------------------------------
__global__ __launch_bounds__(128) void proj_kernel(
    const float* __restrict__ ctx, const float* __restrict__ Wp,
    const float* __restrict__ bp, float* __restrict__ out) {
  const int lane = threadIdx.x & 31;
  const int wave = threadIdx.x >> 5;
  const int half = lane >> 4;
  const int lc   = lane & 15;
  const int koff = half * 2;
  const int m0   = blockIdx.x * 64 + wave * 16;
  const int n0   = blockIdx.y * 64;
  const int arow = m0 + lc;

  v8f acc[4] = {};
  for (int kb = 0; kb < D_; kb += 4) {
    v2f a = *(const v2f*)&ctx[(size_t)arow * D_ + kb + koff];
#pragma unroll
    for (int t = 0; t < 4; ++t) {
      int col = n0 + t * 16 + lc;
      v2f b;
      b[0] = Wp[(size_t)(kb + koff) * D_ + col];
      b[1] = Wp[(size_t)(kb + koff + 1) * D_ + col];
      acc[t] = wmma4(a, b, acc[t]);
    }
  }
#pragma unroll
  for (int t = 0; t < 4; ++t) {
    int col = n0 + t * 16 + lc;
    float bias = bp[col];
#pragma unroll
    for (int i = 0; i < 8; ++i) {
      int grow = m0 + i + half * 8;
      out[(size_t)grow * D_ + col] = acc[t][i] + bias;
    }
  }
}

// ---------------------------------------------------------------------------
extern "C" void kernel_launch(void* const* d_in, const int* in_sizes, int n_in,
                              void* d_out, int out_size, void* d_ws,
                              size_t ws_size, hipStream_t stream) {
  const float* q            = (const float*)d_in[0];
  // d_in[1] (k) and d_in[2] (v) are ignored by the reference module
  const unsigned char* mask = (const unsigned char*)d_in[3];
  const float* Wqkv         = (const float*)d_in[4];
  const float* bqkv         = (const float*)d_in[5];
  const float* Wp           = (const float*)d_in[6];
  const float* bp           = (const float*)d_in[7];

  float* out  = (float*)d_out;
  float* attn = out + (size_t)B_ * L_ * D_;   // second output, concatenated

  const size_t per = (size_t)B_ * H_ * L_ * DK_;  // 2M floats
  float* Qw = (float*)d_ws;
  float* Kw = Qw + per;
  float* Vw = Kw + per;
  float* Cw = Vw + per;   // ctx [B, L, D]

  dim3 blk(128);
  qkv_kernel<<<dim3(64, 24), blk, 0, stream>>>(q, Wqkv, bqkv, Qw, Kw, Vw);
  attn_kernel<<<dim3(16, 32), blk, 0, stream>>>(Qw, Kw, Vw, mask, attn, Cw);
  proj_kernel<<<dim3(64, 8), blk, 0, stream>>>(Cw, Wp, bp, out);
}